// MaxLikelihoodMergeEncoder_41154376630876
// MI455X (gfx1250) — compile-verified
//
#include <hip/hip_runtime.h>
#include <hip/hip_bf16.h>
#include <math.h>

// Problem constants (from the reference)
#define V_N 1024
#define S_N 256
#define C_N 16
#define A_N 4
#define H_N 64
#define EPSC 1e-12f
#define LRC 0.1f

typedef __attribute__((ext_vector_type(2))) float v2f;
typedef __attribute__((ext_vector_type(8))) float v8f;

// ---------------------------------------------------------------------------
// Kernel 1: precompute site_h[s][h] = b1[h] + sum_c sp[s][c] * W1[2+c][h]
// (V-invariant part of the decoder's first layer; 256x64 floats in d_ws)
// ---------------------------------------------------------------------------
__global__ void site_h_kernel(const float* __restrict__ sp,
                              const float* __restrict__ W1,
                              const float* __restrict__ b1,
                              float* __restrict__ site_h) {
  const int s = blockIdx.x;     // 0..255
  const int h = threadIdx.x;    // 0..63
  float acc = b1[h];
#pragma unroll
  for (int c = 0; c < C_N; ++c)
    acc = fmaf(sp[s * C_N + c], W1[(2 + c) * H_N + h], acc);
  site_h[s * H_N + h] = acc;
}

// ---------------------------------------------------------------------------
// 4x4 helpers (fully unrolled, register-resident)
// ---------------------------------------------------------------------------
__device__ __forceinline__ void mm4(const float* __restrict__ A,
                                    const float* __restrict__ B,
                                    float* __restrict__ C) {
#pragma unroll
  for (int i = 0; i < 4; ++i)
#pragma unroll
    for (int j = 0; j < 4; ++j) {
      float acc = A[i * 4 + 0] * B[0 * 4 + j];
#pragma unroll
      for (int k = 1; k < 4; ++k) acc = fmaf(A[i * 4 + k], B[k * 4 + j], acc);
      C[i * 4 + j] = acc;
    }
}

// P = expm(Q*t), QP = Q @ P   (scaling & squaring, order-9 Taylor Horner)
__device__ __forceinline__ void expm4(const float* __restrict__ Q, float t,
                                      float* __restrict__ P,
                                      float* __restrict__ QP) {
  float A[16];
#pragma unroll
  for (int i = 0; i < 16; ++i) A[i] = Q[i] * t;
  float nrm = 0.f;
#pragma unroll
  for (int i = 0; i < 4; ++i) {
    float r = fabsf(A[i * 4 + 0]) + fabsf(A[i * 4 + 1]) +
              fabsf(A[i * 4 + 2]) + fabsf(A[i * 4 + 3]);
    nrm = fmaxf(nrm, r);
  }
  int sq = 0;
  if (nrm > 0.25f) {
    sq = (int)ceilf(log2f(nrm * 4.0f));
    if (sq < 0) sq = 0;
    if (sq > 30) sq = 30;
  }
  const float scl = exp2f(-(float)sq);
#pragma unroll
  for (int i = 0; i < 16; ++i) A[i] *= scl;

  float Pm[16];
#pragma unroll
  for (int i = 0; i < 16; ++i) Pm[i] = (i % 5 == 0) ? 1.f : 0.f;
#pragma unroll
  for (int k = 9; k >= 1; --k) {
    float T[16];
    mm4(A, Pm, T);
    const float inv = 1.0f / (float)k;
#pragma unroll
    for (int i = 0; i < 16; ++i)
      Pm[i] = ((i % 5 == 0) ? 1.f : 0.f) + T[i] * inv;
  }
  for (int r = 0; r < sq; ++r) {
    float T[16];
    mm4(Pm, Pm, T);
#pragma unroll
    for (int i = 0; i < 16; ++i) Pm[i] = T[i];
  }
#pragma unroll
  for (int i = 0; i < 16; ++i) P[i] = Pm[i];
  mm4(Q, P, QP);  // d/dt expm(tQ) = Q expm(tQ)
}

__device__ __forceinline__ float softplusf(float x) {
  return fmaxf(x, 0.f) + log1pf(expf(-fabsf(x)));
}
__device__ __forceinline__ float clampf(float x, float lo, float hi) {
  return fminf(fmaxf(x, lo), hi);
}

// ---------------------------------------------------------------------------
// Kernel 2: one block per v. 256 threads (8 wave32), thread s = site.
// 3 fused optimizer iterations (decoder GEMM via v_wmma_f32_16x16x4_f32,
// per-site expm + analytic grad, deterministic LDS tree reduction),
// then emit parents from the final (alpha,beta).
// ---------------------------------------------------------------------------
__global__ void __launch_bounds__(256)
merge_encoder_kernel(const float* __restrict__ c1g, const float* __restrict__ c2g,
                     const float* __restrict__ lf1, const float* __restrict__ lf2,
                     const float* __restrict__ W1,  const float* __restrict__ Wq,
                     const float* __restrict__ bq,  const float* __restrict__ Wp,
                     const float* __restrict__ bp,  const float* __restrict__ site_h,
                     float* __restrict__ out) {
  __shared__ float sW2t[32 * 64];      // [n][k]  cols 0-15: Wq, 16-19: Wp, rest 0
  __shared__ float sPC[64];            // parents . W1[0:2,:]
  __shared__ float sOut[256 * 33];     // GEMM output, padded rows (bank spread)
  __shared__ float sRed[512];          // tree reduction (g1, g2)

  const int t = threadIdx.x;
  const int v = blockIdx.x;
  const int lane = t & 31;
  const int wid = t >> 5;
  const int half = lane >> 4;   // 0/1
  const int l16 = lane & 15;

  // Stage combined second-layer weights, transposed [n][k] for 8B K-loads.
  for (int i = t; i < 32 * 64; i += 256) {
    const int n = i >> 6, k = i & 63;
    float val = 0.f;
    if (n < 16)      val = Wq[k * 16 + n];
    else if (n < 20) val = Wp[k * 4 + (n - 16)];
    sW2t[i] = val;
  }

  // Per-v setup (all threads redundantly; uniform scalar work).
  const float c10 = c1g[v * 2 + 0], c11 = c1g[v * 2 + 1];
  const float c20 = c2g[v * 2 + 0], c21 = c2g[v * 2 + 1];
  const float pn = sqrtf(c10 * c10 + c11 * c11 + EPSC);
  const float qn = sqrtf(c20 * c20 + c21 * c21 + EPSC);
  float p0 = c10, p1 = c11, q0 = c20, q1 = c21;
  if (pn > qn) { p0 = c10 * qn / pn; p1 = c11 * qn / pn; }
  if (qn > pn) { q0 = c20 * pn / qn; q1 = c21 * pn / qn; }

  // Felsenstein inputs for this thread's site.
  float lfA[4], lfB[4];
  {
    const float4 x = reinterpret_cast<const float4*>(lf1)[(size_t)v * S_N + t];
    lfA[0] = x.x; lfA[1] = x.y; lfA[2] = x.z; lfA[3] = x.w;
    const float4 y = reinterpret_cast<const float4*>(lf2)[(size_t)v * S_N + t];
    lfB[0] = y.x; lfB[1] = y.y; lfB[2] = y.z; lfB[3] = y.w;
  }
  float bqv[16], bpv[4];
#pragma unroll
  for (int n = 0; n < 16; ++n) bqv[n] = bq[n];
#pragma unroll
  for (int n = 0; n < 4; ++n) bpv[n] = bp[n];

  float alpha = 0.5f, beta = 0.5f;
  __syncthreads();

  for (int it = 0; it < 3; ++it) {
    const float mid0 = p0 * alpha + q0 * (1.f - alpha);
    const float mid1 = p1 * alpha + q1 * (1.f - alpha);
    const float par0 = mid0 * beta, par1 = mid1 * beta;

    if (t < 64) sPC[t] = fmaf(par0, W1[t], par1 * W1[64 + t]);
    __syncthreads();

    // ---- WMMA decoder GEMM: relu(site_h+pc)[256x64] @ W2[64x32] -> sOut ----
    // EXEC all-ones: every lane of every wave runs this region, no divergence.
#pragma unroll
    for (int mi = 0; mi < 2; ++mi) {
      const int mt = wid * 2 + mi;
      const int row = mt * 16 + l16;                 // site index (A-matrix M)
      const float* hrow = site_h + row * H_N;
#pragma unroll
      for (int nt = 0; nt < 2; ++nt) {
        v8f acc = {0.f, 0.f, 0.f, 0.f, 0.f, 0.f, 0.f, 0.f};
#pragma unroll
        for (int kc = 0; kc < 16; ++kc) {
          const int k = kc * 4 + half * 2;           // f32 16x4 A layout
          v2f a, b;
          a.x = fmaxf(hrow[k + 0] + sPC[k + 0], 0.f);
          a.y = fmaxf(hrow[k + 1] + sPC[k + 1], 0.f);
          b.x = sW2t[(nt * 16 + l16) * 64 + k + 0];
          b.y = sW2t[(nt * 16 + l16) * 64 + k + 1];
          acc = __builtin_amdgcn_wmma_f32_16x16x4_f32(
              false, a, false, b, (short)0, acc, false, false);
        }
#pragma unroll
        for (int g = 0; g < 8; ++g) {
          const int orow = mt * 16 + g + 8 * half;   // 16x16 f32 C/D layout
          sOut[orow * 33 + nt * 16 + l16] = acc[g];
        }
      }
    }
    __syncthreads();

    // ---- per-site epilogue: Q, stat, expm, analytic d ll/d branch ----
    float Qm[16];
    {
      float rates[16];
#pragma unroll
      for (int n = 0; n < 16; ++n)
        rates[n] = softplusf(sOut[t * 33 + n] + bqv[n]);
#pragma unroll
      for (int i = 0; i < 4; ++i) {
        float rs = 0.f;
#pragma unroll
        for (int j = 0; j < 4; ++j)
          if (j != i) rs += rates[i * 4 + j];
#pragma unroll
        for (int j = 0; j < 4; ++j)
          Qm[i * 4 + j] = (i == j) ? -rs : rates[i * 4 + j];
      }
    }
    float lstat[4];
    {
      float pp[4], mx = -1e30f;
#pragma unroll
      for (int j = 0; j < 4; ++j) {
        pp[j] = sOut[t * 33 + 16 + j] + bpv[j];
        mx = fmaxf(mx, pp[j]);
      }
      float se = 0.f;
#pragma unroll
      for (int j = 0; j < 4; ++j) se += expf(pp[j] - mx);
      const float lse = mx + logf(se);
#pragma unroll
      for (int j = 0; j < 4; ++j) lstat[j] = pp[j] - lse;
    }

    const float e10 = par0 - c10, e11 = par1 - c11;
    const float e20 = par0 - c20, e21 = par1 - c21;
    const float t1 = sqrtf(e10 * e10 + e11 * e11 + EPSC);
    const float t2 = sqrtf(e20 * e20 + e21 * e21 + EPSC);

    float Lc[2][4], Sg[2][4];
#pragma unroll
    for (int c = 0; c < 2; ++c) {
      const float tb = (c == 0) ? t1 : t2;
      const float* lfc = (c == 0) ? lfA : lfB;
      float P[16], QP[16];
      expm4(Qm, tb, P, QP);
#pragma unroll
      for (int a = 0; a < 4; ++a) {
        float lp[4], m = -1e30f;
#pragma unroll
        for (int b = 0; b < 4; ++b) {
          lp[b] = logf(fmaxf(P[a * 4 + b], 1e-30f)) + lfc[b];
          m = fmaxf(m, lp[b]);
        }
        float ssum = 0.f;
#pragma unroll
        for (int b = 0; b < 4; ++b) ssum += expf(lp[b] - m);
        const float Lv = m + logf(ssum);
        Lc[c][a] = Lv;
        float g = 0.f;
#pragma unroll
        for (int b = 0; b < 4; ++b)
          if (P[a * 4 + b] > 1e-30f)               // clip() grad mask
            g = fmaf(expf(lfc[b] - Lv), QP[a * 4 + b], g);
        Sg[c][a] = g;
      }
    }

    float g1 = 0.f, g2 = 0.f;
    {
      float z[4], zm = -1e30f;
#pragma unroll
      for (int a = 0; a < 4; ++a) {
        z[a] = Lc[0][a] + Lc[1][a] + lstat[a];
        zm = fmaxf(zm, z[a]);
      }
      float es[4], ss = 0.f;
#pragma unroll
      for (int a = 0; a < 4; ++a) { es[a] = expf(z[a] - zm); ss += es[a]; }
      const float inv = 1.0f / ss;
#pragma unroll
      for (int a = 0; a < 4; ++a) {
        const float w = es[a] * inv;
        g1 = fmaf(w, Sg[0][a], g1);
        g2 = fmaf(w, Sg[1][a], g2);
      }
    }

    // ---- deterministic tree reduction over sites ----
    sRed[t] = g1;
    sRed[256 + t] = g2;
    __syncthreads();
    for (int off = 128; off > 0; off >>= 1) {
      if (t < off) {
        sRed[t] += sRed[t + off];
        sRed[256 + t] += sRed[256 + t + off];
      }
      __syncthreads();
    }
    const float G1 = sRed[0], G2 = sRed[256];

    // ---- chain rule into (alpha, beta); every thread computes identically ----
    const float dpa0 = beta * (p0 - q0), dpa1 = beta * (p1 - q1);
    const float db1da = (e10 * dpa0 + e11 * dpa1) / t1;
    const float db1db = (e10 * mid0 + e11 * mid1) / t1;
    const float db2da = (e20 * dpa0 + e21 * dpa1) / t2;
    const float db2db = (e20 * mid0 + e21 * mid1) / t2;
    const float ga = G1 * db1da + G2 * db2da;
    const float gb = G1 * db1db + G2 * db2db;
    alpha = clampf(alpha + LRC * clampf(ga, -1.f, 1.f), 0.f, 1.f);
    beta  = clampf(beta  + LRC * clampf(gb, -1.f, 1.f), 0.f, 1.f);
  }

  // Final parents (scan entry 3: computed from alpha/beta after 3 updates).
  if (t == 0) {
    const float mid0 = p0 * alpha + q0 * (1.f - alpha);
    const float mid1 = p1 * alpha + q1 * (1.f - alpha);
    out[v * 2 + 0] = mid0 * beta;
    out[v * 2 + 1] = mid1 * beta;
  }
}

// ---------------------------------------------------------------------------
extern "C" void kernel_launch(void* const* d_in, const int* in_sizes, int n_in,
                              void* d_out, int out_size, void* d_ws, size_t ws_size,
                              hipStream_t stream) {
  (void)in_sizes; (void)n_in; (void)out_size; (void)ws_size;
  const float* c1 = (const float*)d_in[0];
  const float* c2 = (const float*)d_in[1];
  const float* lf1 = (const float*)d_in[2];
  const float* lf2 = (const float*)d_in[3];
  const float* sp = (const float*)d_in[4];
  const float* W1 = (const float*)d_in[5];
  const float* b1 = (const float*)d_in[6];
  const float* Wq = (const float*)d_in[7];
  const float* bq = (const float*)d_in[8];
  const float* Wp = (const float*)d_in[9];
  const float* bp = (const float*)d_in[10];
  float* out = (float*)d_out;
  float* site_h = (float*)d_ws;  // 256*64 floats = 64 KB

  site_h_kernel<<<S_N, H_N, 0, stream>>>(sp, W1, b1, site_h);
  merge_encoder_kernel<<<V_N, 256, 0, stream>>>(c1, c2, lf1, lf2, W1, Wq, bq,
                                                Wp, bp, site_h, out);
}